// PBGNN_16758962389038
// MI455X (gfx1250) — compile-verified
//
#include <hip/hip_runtime.h>
#include <math.h>

// ---------------------------------------------------------------------------
// SchNet-style CFConv for MI455X (gfx1250, wave32, WMMA).
// N=50000, A=F=128, R=20, E=1600000.
//
// All GEMMs via V_WMMA_F32_16X16X4_F32 (fp32-exact; problem is BW/L2-bound).
// Weights are repacked once per launch into paired-K layout so every B
// fragment is a single aligned float2 (global_load_b64); A fragments are
// hoisted into registers once per K-sweep (8x fewer A loads). ssp uses the
// hardware v_exp_f32/v_log_f32 TRANS ops which co-execute with WMMA.
// ---------------------------------------------------------------------------

typedef __attribute__((ext_vector_type(2))) float v2f;
typedef __attribute__((ext_vector_type(8))) float v8f;

#define A_DIM 128
#define F_DIM 128
#define R_DIM 20
#define LDS_STRIDE 132   // 128 + 4 pad: row offset shifts banks by 4
#define WAVES_PER_BLOCK 5
#define BLOCK_THREADS (WAVES_PER_BLOCK * 32)

__device__ __forceinline__ v8f wmma4(v2f a, v2f b, v8f c) {
  // D = A(16x4 f32) * B(4x16 f32) + C(16x16 f32)
  return __builtin_amdgcn_wmma_f32_16x16x4_f32(
      /*neg_a=*/false, a, /*neg_b=*/false, b,
      /*c_mod=*/(short)0, c, /*reuse_a=*/false, /*reuse_b=*/false);
}

__device__ __forceinline__ v8f v8f_zero() {
  v8f z = {0.0f, 0.0f, 0.0f, 0.0f, 0.0f, 0.0f, 0.0f, 0.0f};
  return z;
}

// shifted softplus: softplus(x) - ln(2). Stable: max(x,0)+log(1+exp(-|x|))-ln2.
// exp/log use the hardware TRANS ops (v_exp_f32 / v_log_f32); argument of the
// log is in (1,2] so there is no cancellation regime.
__device__ __forceinline__ float ssp(float x) {
  const float t = __expf(-fabsf(x));
  return fmaxf(x, 0.0f) + __logf(1.0f + t) - 0.69314718055994531f;
}

// ---------------------------------------------------------------------------
// zero-fill workspace accumulator
// ---------------------------------------------------------------------------
__global__ void zero_kernel(float* __restrict__ p, long n) {
  long i = (long)blockIdx.x * blockDim.x + threadIdx.x;
  if (i < n) p[i] = 0.0f;
}

// ---------------------------------------------------------------------------
// Repack row-major W[K][Ncol] into paired-K layout:
//   Wp[k2][n][2] = { W[2*k2][n], W[2*k2+1][n] }   (row length 2*Ncol)
// so a B fragment (elements from rows kb, kb+1 at column n) is one float2.
// ---------------------------------------------------------------------------
__global__ void pack_kernel(const float* __restrict__ W, float* __restrict__ Wp,
                            int K, int Ncol) {
  int i = blockIdx.x * blockDim.x + threadIdx.x;
  int total = (K / 2) * Ncol;
  if (i >= total) return;
  int k2 = i / Ncol;
  int n  = i - k2 * Ncol;
  Wp[k2 * 2 * Ncol + 2 * n + 0] = W[(2 * k2 + 0) * Ncol + n];
  Wp[k2 * 2 * Ncol + 2 * n + 1] = W[(2 * k2 + 1) * Ncol + n];
}

// B-fragment fetch from packed weights: kb = 4*k + 2*khalf -> k2 = 2*k+khalf
__device__ __forceinline__ v2f bfrag(const float* __restrict__ Wp,
                                     int k, int khalf, int n) {
  return *(const v2f*)(Wp + (2 * k + khalf) * 256 + 2 * n);
}

// ---------------------------------------------------------------------------
// h = x @ W_in2f   ([N,128] x [128,128], no bias)
// One wave computes one 16-row tile across all 8 column tiles.
// ---------------------------------------------------------------------------
__global__ void __launch_bounds__(BLOCK_THREADS)
gemm_h_kernel(const float* __restrict__ x, const float* __restrict__ Wp,
              float* __restrict__ h, int ntiles) {
  const int wave  = threadIdx.x >> 5;
  const int lane  = threadIdx.x & 31;
  const int t     = blockIdx.x * WAVES_PER_BLOCK + wave;
  if (t >= ntiles) return;                       // wave-uniform exit
  const int mrow  = lane & 15;                   // row (A) / col (B,C) in tile
  const int khalf = lane >> 4;                   // 0: K0/K1, 1: K2/K3
  const long r0   = (long)t * 16;

  // hoist all A fragments for the K=128 sweep (reused by all 8 col tiles)
  const float* xrow = x + (r0 + mrow) * A_DIM;
  v2f af[32];
#pragma unroll
  for (int k = 0; k < 32; ++k)
    af[k] = *(const v2f*)(xrow + 4 * k + 2 * khalf);

  for (int j = 0; j < 8; ++j) {
    const int n = j * 16 + mrow;
    v8f acc = v8f_zero();
#pragma unroll
    for (int k = 0; k < 32; ++k)
      acc = wmma4(af[k], bfrag(Wp, k, khalf, n), acc);
#pragma unroll
    for (int r = 0; r < 8; ++r) {
      const int m = r + 8 * khalf;                // C-layout row
      h[(r0 + m) * F_DIM + n] = acc[r];
    }
  }
}

// ---------------------------------------------------------------------------
// Fused edge pipeline, one wave per 16-edge tile:
//   t1  = ssp(f_ij[tile] @ W_f1 + b_f1)          (K=20, 5 k-steps)   -> LDS
//   Wij = (t1 @ W_f2 + b_f2) * rcut              (K=128, 256 WMMAs)
//   agg[idx_i] += h[idx_j] * Wij                 (fused gather/scatter)
// ---------------------------------------------------------------------------
__global__ void __launch_bounds__(BLOCK_THREADS)
edge_kernel(const float* __restrict__ f_ij, const float* __restrict__ rcut,
            const int* __restrict__ idx_i, const int* __restrict__ idx_j,
            const float* __restrict__ Wp_f1, const float* __restrict__ b_f1,
            const float* __restrict__ Wp_f2, const float* __restrict__ b_f2,
            const float* __restrict__ h, float* __restrict__ agg, int ntiles) {
  __shared__ float lds[WAVES_PER_BLOCK][16 * LDS_STRIDE];

  const int wave   = threadIdx.x >> 5;
  const int lane   = threadIdx.x & 31;
  const int t      = blockIdx.x * WAVES_PER_BLOCK + wave;
  const bool live  = (t < ntiles);               // wave-uniform
  const int mrow   = lane & 15;
  const int khalf  = lane >> 4;
  const long e0    = (long)t * 16;
  float* tw = lds[wave];

  if (live) {
    // ---- GEMM1: [16 x 20] @ [20 x 128] -> acc1 (C-layout, 8 col tiles) ----
    const float* frow = f_ij + (e0 + mrow) * R_DIM;
    v2f af1[5];
#pragma unroll
    for (int k = 0; k < 5; ++k)                   // R = 20 = 5 * 4 exactly
      af1[k] = *(const v2f*)(frow + 4 * k + 2 * khalf);

    v8f acc1[8];
#pragma unroll
    for (int j = 0; j < 8; ++j) acc1[j] = v8f_zero();
#pragma unroll
    for (int k = 0; k < 5; ++k) {
#pragma unroll
      for (int j = 0; j < 8; ++j)
        acc1[j] = wmma4(af1[k], bfrag(Wp_f1, k, khalf, j * 16 + mrow), acc1[j]);
    }
    // ---- bias + shifted-softplus, stage activation tile into LDS ----
#pragma unroll
    for (int j = 0; j < 8; ++j) {
      const int n = j * 16 + mrow;
      const float bias = b_f1[n];
#pragma unroll
      for (int r = 0; r < 8; ++r) {
        const int m = r + 8 * khalf;
        tw[m * LDS_STRIDE + n] = ssp(acc1[j][r] + bias);
      }
    }
  }
  __syncthreads();                                // uniform across block

  if (live) {
    // per-row edge metadata (C-layout row r + 8*khalf), hoisted once
    int   vi[8]; int vj[8]; float vr[8];
#pragma unroll
    for (int r = 0; r < 8; ++r) {
      const long e = e0 + r + 8 * khalf;
      vi[r] = idx_i[e]; vj[r] = idx_j[e]; vr[r] = rcut[e];
    }

    // hoist all 32 A fragments of the activation tile from LDS (reused 8x)
    const float* arow = tw + mrow * LDS_STRIDE;
    v2f af2[32];
#pragma unroll
    for (int k = 0; k < 32; ++k)
      af2[k] = *(const v2f*)(arow + 4 * k + 2 * khalf);

    // ---- GEMM2: [16 x 128] @ [128 x 128] + fused epilogue ----
    for (int j2 = 0; j2 < 8; ++j2) {
      const int n = j2 * 16 + mrow;
      v8f acc2 = v8f_zero();
#pragma unroll
      for (int k = 0; k < 32; ++k)
        acc2 = wmma4(af2[k], bfrag(Wp_f2, k, khalf, n), acc2);

      const float bias2 = b_f2[n];
#pragma unroll
      for (int r = 0; r < 8; ++r) {
        const float wij = (acc2[r] + bias2) * vr[r];        // filter * cutoff
        const float xv  = h[(long)vj[r] * F_DIM + n];       // gather (L2-hot)
        atomicAdd(agg + (long)vi[r] * F_DIM + n, wij * xv); // scatter-add
      }
    }
  }
}

// ---------------------------------------------------------------------------
// out = ssp(agg @ W_o1 + b_o1) @ W_o2 + b_o2   (fused two-GEMM, N rows)
// ---------------------------------------------------------------------------
__global__ void __launch_bounds__(BLOCK_THREADS)
out_kernel(const float* __restrict__ agg,
           const float* __restrict__ Wp_o1, const float* __restrict__ b_o1,
           const float* __restrict__ Wp_o2, const float* __restrict__ b_o2,
           float* __restrict__ out, int ntiles) {
  __shared__ float lds[WAVES_PER_BLOCK][16 * LDS_STRIDE];

  const int wave  = threadIdx.x >> 5;
  const int lane  = threadIdx.x & 31;
  const int t     = blockIdx.x * WAVES_PER_BLOCK + wave;
  const bool live = (t < ntiles);
  const int mrow  = lane & 15;
  const int khalf = lane >> 4;
  const long r0   = (long)t * 16;
  float* tw = lds[wave];

  if (live) {
    const float* grow = agg + (r0 + mrow) * F_DIM;
    v2f af[32];
#pragma unroll
    for (int k = 0; k < 32; ++k)
      af[k] = *(const v2f*)(grow + 4 * k + 2 * khalf);

    for (int j = 0; j < 8; ++j) {
      const int n = j * 16 + mrow;
      v8f acc = v8f_zero();
#pragma unroll
      for (int k = 0; k < 32; ++k)
        acc = wmma4(af[k], bfrag(Wp_o1, k, khalf, n), acc);

      const float bias = b_o1[n];
#pragma unroll
      for (int r = 0; r < 8; ++r) {
        const int m = r + 8 * khalf;
        tw[m * LDS_STRIDE + n] = ssp(acc[r] + bias);
      }
    }
  }
  __syncthreads();

  if (live) {
    const float* arow = tw + mrow * LDS_STRIDE;
    v2f af2[32];
#pragma unroll
    for (int k = 0; k < 32; ++k)
      af2[k] = *(const v2f*)(arow + 4 * k + 2 * khalf);

    for (int j2 = 0; j2 < 8; ++j2) {
      const int n = j2 * 16 + mrow;
      v8f acc = v8f_zero();
#pragma unroll
      for (int k = 0; k < 32; ++k)
        acc = wmma4(af2[k], bfrag(Wp_o2, k, khalf, n), acc);

      const float bias2 = b_o2[n];
#pragma unroll
      for (int r = 0; r < 8; ++r) {
        const int m = r + 8 * khalf;
        out[(r0 + m) * A_DIM + n] = acc[r] + bias2;
      }
    }
  }
}

// ---------------------------------------------------------------------------
extern "C" void kernel_launch(void* const* d_in, const int* in_sizes, int n_in,
                              void* d_out, int out_size, void* d_ws, size_t ws_size,
                              hipStream_t stream) {
  const float* x      = (const float*)d_in[0];
  const float* f_ij   = (const float*)d_in[1];
  const float* rcut   = (const float*)d_in[2];
  const int*   idx_i  = (const int*)d_in[3];
  const int*   idx_j  = (const int*)d_in[4];
  const float* W_in2f = (const float*)d_in[5];
  const float* W_f1   = (const float*)d_in[6];
  const float* b_f1   = (const float*)d_in[7];
  const float* W_f2   = (const float*)d_in[8];
  const float* b_f2   = (const float*)d_in[9];
  const float* W_o1   = (const float*)d_in[10];
  const float* b_o1   = (const float*)d_in[11];
  const float* W_o2   = (const float*)d_in[12];
  const float* b_o2   = (const float*)d_in[13];
  float* out = (float*)d_out;

  const int  N = in_sizes[0] / A_DIM;   // 50000
  const long E = (long)in_sizes[2];     // 1600000 (rcut_ij length)

  float* h       = (float*)d_ws;                     // [N, F]
  float* agg     = h + (size_t)N * F_DIM;            // [N, F]
  float* Wp_in2f = agg + (size_t)N * F_DIM;          // [A/2, 2F]
  float* Wp_f1   = Wp_in2f + A_DIM * F_DIM;          // [R/2, 2F]
  float* Wp_f2   = Wp_f1 + R_DIM * F_DIM;            // [F/2, 2F]
  float* Wp_o1   = Wp_f2 + F_DIM * F_DIM;            // [F/2, 2A]
  float* Wp_o2   = Wp_o1 + F_DIM * A_DIM;            // [A/2, 2A]

  // repack weights into paired-K layout (one float2 per B fragment)
  {
    int blk = 256;
    pack_kernel<<<(A_DIM / 2 * F_DIM + blk - 1) / blk, blk, 0, stream>>>(W_in2f, Wp_in2f, A_DIM, F_DIM);
    pack_kernel<<<(R_DIM / 2 * F_DIM + blk - 1) / blk, blk, 0, stream>>>(W_f1, Wp_f1, R_DIM, F_DIM);
    pack_kernel<<<(F_DIM / 2 * F_DIM + blk - 1) / blk, blk, 0, stream>>>(W_f2, Wp_f2, F_DIM, F_DIM);
    pack_kernel<<<(F_DIM / 2 * A_DIM + blk - 1) / blk, blk, 0, stream>>>(W_o1, Wp_o1, F_DIM, A_DIM);
    pack_kernel<<<(A_DIM / 2 * A_DIM + blk - 1) / blk, blk, 0, stream>>>(W_o2, Wp_o2, A_DIM, A_DIM);
  }
  // zero the scatter accumulator
  {
    long n = (long)N * F_DIM;
    int  blk = 256;
    zero_kernel<<<(int)((n + blk - 1) / blk), blk, 0, stream>>>(agg, n);
  }
  // h = x @ W_in2f
  {
    int ntiles = N / 16;                                    // 3125
    int blocks = (ntiles + WAVES_PER_BLOCK - 1) / WAVES_PER_BLOCK;
    gemm_h_kernel<<<blocks, BLOCK_THREADS, 0, stream>>>(x, Wp_in2f, h, ntiles);
  }
  // fused edge filter + gather + scatter
  {
    int ntiles = (int)(E / 16);                             // 100000
    int blocks = (ntiles + WAVES_PER_BLOCK - 1) / WAVES_PER_BLOCK;
    edge_kernel<<<blocks, BLOCK_THREADS, 0, stream>>>(
        f_ij, rcut, idx_i, idx_j, Wp_f1, b_f1, Wp_f2, b_f2, h, agg, ntiles);
  }
  // out = ssp(agg @ W_o1 + b_o1) @ W_o2 + b_o2
  {
    int ntiles = N / 16;
    int blocks = (ntiles + WAVES_PER_BLOCK - 1) / WAVES_PER_BLOCK;
    out_kernel<<<blocks, BLOCK_THREADS, 0, stream>>>(
        agg, Wp_o1, b_o1, Wp_o2, b_o2, out, ntiles);
  }
}